// AlphaPermutationLayer_18451179503707
// MI455X (gfx1250) — compile-verified
//
#include <hip/hip_runtime.h>
#include <math.h>

#define PN 2048   // permutation size (output is PN x PN)
#define PK 64     // pool size

// Raw 32-bit LDS byte offset of a pointer that is known to point into LDS.
// generic -> address_space(3) addrspacecast strips the LDS aperture base.
__device__ __forceinline__ unsigned lds_byte_off(const void* p) {
    return (unsigned)(unsigned long long)
        (__attribute__((address_space(3))) const char*)p;
}

__global__ void __launch_bounds__(256)
alpha_perm_scatter_kernel(const float* __restrict__ alpha,
                          const long long* __restrict__ perm,  // [PK, PN] int64
                          const float* __restrict__ temp,      // scalar
                          float* __restrict__ out)             // [PN, PN] f32
{
    __shared__ __align__(16) float row[PN];   // 8 KB staging for one output row
    __shared__ float logit[PK];

    const int i = blockIdx.x;    // output row index
    const int t = threadIdx.x;   // 0..255

    // 1) scaled logits into LDS
    if (t < PK) logit[t] = alpha[t] / temp[0];

    // 2) zero the row staging buffer with wide LDS stores (2 x b128 / thread)
    float4* row4 = (float4*)row;
    const float4 z = make_float4(0.f, 0.f, 0.f, 0.f);
    row4[t]       = z;
    row4[t + 256] = z;

    __syncthreads();

    // 3) softmax (redundant per-thread over 64 LDS values: trivial) + scatter
    if (t < PK) {
        float m = -INFINITY;
        #pragma unroll
        for (int j = 0; j < PK; ++j) m = fmaxf(m, logit[j]);
        float s = 0.f;
        #pragma unroll
        for (int j = 0; j < PK; ++j) s += __expf(logit[j] - m);
        const float a = __expf(logit[t] - m) / s;

        // perm_vectors[k, i] at flat index k*PN + i (int64, value < 2048)
        const int col = (int)perm[(size_t)t * PN + i];
        atomicAdd(&row[col], a);   // ds_add_f32: resolves k-collisions per row
    }

    __syncthreads();

    // 4) drain the finished row to HBM via the CDNA5 async DMA path:
    //    per-lane LDS -> global 128-bit async stores, tracked on ASYNCcnt.
    const unsigned long long gbase =
        (unsigned long long)(out + (size_t)i * PN);
    #pragma unroll
    for (int half = 0; half < 2; ++half) {
        const int col = half * 1024 + t * 4;           // 4 floats per lane
        const unsigned long long gaddr = gbase + (unsigned long long)col * 4ull;
        const unsigned laddr = lds_byte_off(&row[col]);
        asm volatile("global_store_async_from_lds_b128 %0, %1, off"
                     :: "v"(gaddr), "v"(laddr)
                     : "memory");
    }
    asm volatile("s_wait_asynccnt 0" ::: "memory");
}

extern "C" void kernel_launch(void* const* d_in, const int* in_sizes, int n_in,
                              void* d_out, int out_size, void* d_ws, size_t ws_size,
                              hipStream_t stream) {
    (void)in_sizes; (void)n_in; (void)d_ws; (void)ws_size; (void)out_size;

    const float*     alpha = (const float*)d_in[0];      // [64] f32
    const long long* perm  = (const long long*)d_in[1];  // [64, 2048] i64
    const float*     temp  = (const float*)d_in[2];      // [1] f32
    float*           out   = (float*)d_out;              // [2048, 2048] f32

    alpha_perm_scatter_kernel<<<PN, 256, 0, stream>>>(alpha, perm, temp, out);
}